// MfdFC_42039139894261
// MI455X (gfx1250) — compile-verified
//
#include <hip/hip_runtime.h>

typedef float v2f __attribute__((ext_vector_type(2)));
typedef float v8f __attribute__((ext_vector_type(8)));

namespace {
constexpr int   C    = 64;   // C_in == C_out == d == 64
constexpr int   STR  = 68;   // padded LDS row stride (floats): bank shift 4/row
constexpr float kEps = 1e-7f;
}

// Column softmax-normalize w_raw -> wn (shared by all tokens; 16 KB, L2-resident).
__global__ __launch_bounds__(64)
void mfdfc_wnorm_kernel(const float* __restrict__ w_raw, float* __restrict__ wn) {
  const int o = threadIdx.x;
  const float lc = 4.1588830833596715f;  // ln(64)
  float s = 0.0f;
  for (int i = 0; i < C; ++i) s += expf(w_raw[i * C + o] - lc);
  const float inv = 1.0f / s;
  for (int i = 0; i < C; ++i) wn[i * C + o] = expf(w_raw[i * C + o] - lc) * inv;
}

// Deterministic tree sum over each 16-lane half of a wave32.
__device__ __forceinline__ float half16_sum(float v) {
  v += __shfl_xor(v, 1, 32);
  v += __shfl_xor(v, 2, 32);
  v += __shfl_xor(v, 4, 32);
  v += __shfl_xor(v, 8, 32);
  return v;
}

__global__ __launch_bounds__(256)
void mfdfc_main_kernel(const float* __restrict__ x, const float* __restrict__ wn,
                       float* __restrict__ out) {
  __shared__ float xs[C * STR];     // x tile for this token: xs[i][k]
  __shared__ float as[C * STR];     // current means: as[o][k]
  __shared__ float ss[C * STR];     // S matrix: ss[o][i]
  __shared__ float tpart[C * 4];    // partial sums of s*dot per (o, i_tile)
  __shared__ float gpart[C * 4];    // partial sums of grad^2 per (o, k_tile)
  __shared__ float pa[C];           // ||a_o||^2
  __shared__ float qq[C];           // ||x_i||^2
  __shared__ float tt[C];           // sum_i s*dot per o
  __shared__ float cg[C];           // cos(||grad_o||)
  __shared__ float sg[C];           // sinc(||grad_o||)

  const int tid  = threadIdx.x;
  const int lane = tid & 31;
  const int wave = tid >> 5;
  const int il   = lane & 15;   // N-index within 16x16 tile
  const int hh   = lane >> 4;   // half-wave (selects K pair / M+8)
  const int n    = blockIdx.x;

  const float* xn = x + (size_t)n * C * C;

  // ---- stage x tile into LDS (coalesced float4) ----
  for (int j = 0; j < 4; ++j) {
    int f4  = tid + 256 * j;        // 1024 float4 chunks = 64x64 floats
    int row = f4 >> 4;
    int c4  = f4 & 15;
    const float4 v = ((const float4*)xn)[f4];
    *(float4*)&xs[row * STR + 4 * c4] = v;
  }
  __syncthreads();

  // qq[i] = ||x_i||^2 (once); init means a_o = x_0 for all o
  if (tid < C) {
    float s = 0.0f;
    for (int k = 0; k < C; ++k) { float v = xs[tid * STR + k]; s += v * v; }
    qq[tid] = s;
  }
  for (int j = 0; j < 16; ++j) {
    int f = tid + 256 * j;
    int o = f >> 6, k = f & 63;
    as[o * STR + k] = xs[k];        // broadcast row 0 of x
  }

  // Per-lane w values for this wave's two tiles (constant across iterations)
  float wreg[2][8];
  for (int u = 0; u < 2; ++u) {
    int t  = wave * 2 + u;
    int m0 = (t >> 2) * 16;
    int i0 = (t & 3) * 16;
    int i  = i0 + il;
    for (int r = 0; r < 8; ++r) {
      int o = m0 + r + 8 * hh;
      wreg[u][r] = wn[i * C + o];
    }
  }
  __syncthreads();

  for (int it = 0; it < 3; ++it) {
    // pa[o] = ||a_o||^2 (a changes every iteration)
    if (tid < C) {
      float s = 0.0f;
      for (int k = 0; k < C; ++k) { float v = as[tid * STR + k]; s += v * v; }
      pa[tid] = s;
    }
    __syncthreads();

    // ---- GEMM 1 (WMMA f32): dot[o,i] = a[o,:] . x[i,:]; then log-map scalars ----
    for (int u = 0; u < 2; ++u) {
      int t  = wave * 2 + u;
      int m0 = (t >> 2) * 16;   // o tile
      int i0 = (t & 3) * 16;    // i tile
      v8f acc = {};
      for (int kc = 0; kc < C; kc += 4) {
        // A frag: lane holds a[m0+il][kc+2h .. +1] ; B frag: x[i0+il][kc+2h .. +1]
        v2f af = *(const v2f*)&as[(m0 + il) * STR + kc + 2 * hh];
        v2f bf = *(const v2f*)&xs[(i0 + il) * STR + kc + 2 * hh];
        acc = __builtin_amdgcn_wmma_f32_16x16x4_f32(false, af, false, bf,
                                                    (short)0, acc, false, false);
      }
      int i = i0 + il;
      for (int r = 0; r < 8; ++r) {
        int o = m0 + r + 8 * hh;
        float du  = acc[r];
        float dc  = fminf(fmaxf(du, -1.0f + kEps), 1.0f - kEps);
        float th  = acosf(dc);
        float nv2 = qq[i] - 2.0f * dc * du + dc * dc * pa[o];
        float nv  = sqrtf(fmaxf(nv2, 0.0f));
        float fac = (nv < kEps) ? 1.0f : th / nv;
        float s   = wreg[u][r] * fac;
        ss[o * STR + i] = s;
        float part = half16_sum(s * dc);
        if (il == 0) tpart[o * 4 + (t & 3)] = part;
      }
    }
    __syncthreads();

    if (tid < C)
      tt[tid] = tpart[tid * 4 + 0] + tpart[tid * 4 + 1] +
                tpart[tid * 4 + 2] + tpart[tid * 4 + 3];
    __syncthreads();

    // ---- GEMM 2 (WMMA f32): g0[o,k] = sum_i s[o,i] x[i,k]; grad = g0 - tt[o]*a ----
    float greg[2][8];
    for (int u = 0; u < 2; ++u) {
      int t  = wave * 2 + u;
      int o0 = (t >> 2) * 16;   // o tile
      int k0 = (t & 3) * 16;    // k tile
      v8f acc = {};
      for (int ic = 0; ic < C; ic += 4) {
        v2f af = *(const v2f*)&ss[(o0 + il) * STR + ic + 2 * hh];
        v2f bf;
        bf.x = xs[(ic + 2 * hh)     * STR + k0 + il];
        bf.y = xs[(ic + 2 * hh + 1) * STR + k0 + il];
        acc = __builtin_amdgcn_wmma_f32_16x16x4_f32(false, af, false, bf,
                                                    (short)0, acc, false, false);
      }
      int k = k0 + il;
      for (int r = 0; r < 8; ++r) {
        int o = o0 + r + 8 * hh;
        float g = acc[r] - tt[o] * as[o * STR + k];
        greg[u][r] = g;
        float part = half16_sum(g * g);
        if (il == 0) gpart[o * 4 + (t & 3)] = part;
      }
    }
    __syncthreads();

    if (tid < C) {
      float s2 = gpart[tid * 4 + 0] + gpart[tid * 4 + 1] +
                 gpart[tid * 4 + 2] + gpart[tid * 4 + 3];
      float gl = sqrtf(s2);
      cg[tid] = cosf(gl);
      sg[tid] = (gl < kEps) ? 1.0f : (sinf(gl) / gl);
    }
    __syncthreads();

    // ---- exp-map update: a = cos(|g|)*a + sinc(|g|)*grad ----
    for (int u = 0; u < 2; ++u) {
      int t  = wave * 2 + u;
      int o0 = (t >> 2) * 16;
      int k0 = (t & 3) * 16;
      int k  = k0 + il;
      for (int r = 0; r < 8; ++r) {
        int o = o0 + r + 8 * hh;
        as[o * STR + k] = cg[o] * as[o * STR + k] + sg[o] * greg[u][r];
      }
    }
    __syncthreads();
  }

  // ---- write result (coalesced) ----
  float* on = out + (size_t)n * C * C;
  for (int j = 0; j < 16; ++j) {
    int f = tid + 256 * j;
    int o = f >> 6, k = f & 63;
    on[f] = as[o * STR + k];
  }
}

extern "C" void kernel_launch(void* const* d_in, const int* in_sizes, int n_in,
                              void* d_out, int out_size, void* d_ws, size_t ws_size,
                              hipStream_t stream) {
  (void)in_sizes; (void)n_in; (void)out_size; (void)ws_size;
  const float* x     = (const float*)d_in[0];   // (4,256,64,64) f32
  const float* w_raw = (const float*)d_in[1];   // (64,64) f32
  float* out = (float*)d_out;                   // (4,256,64,64) f32
  float* wn  = (float*)d_ws;                    // normalized w scratch (4096 f32)

  mfdfc_wnorm_kernel<<<1, 64, 0, stream>>>(w_raw, wn);
  mfdfc_main_kernel<<<1024, 256, 0, stream>>>(x, wn, out);
}